// GNN_39840116638112
// MI455X (gfx1250) — compile-verified
//
#include <hip/hip_runtime.h>

#define D 64

typedef __attribute__((ext_vector_type(2))) float v2f;
typedef __attribute__((ext_vector_type(8))) float v8f;

// ---------------------------------------------------------------------------
// Zero a float buffer (graph-capture-safe replacement for hipMemsetAsync)
// ---------------------------------------------------------------------------
__global__ void zero_f32(float* __restrict__ p, long n) {
  long i = (long)blockIdx.x * blockDim.x + threadIdx.x;
  long stride = (long)gridDim.x * blockDim.x;
  for (; i < n; i += stride) p[i] = 0.0f;
}

// ---------------------------------------------------------------------------
// Scatter-add of messages: ssum[dst] += x[src], cnt[dst] += 1 (optional).
// One work item per (edge, float4 chunk): E*16 items, each does a 16B load
// and 4 global float atomics (L2-resident on MI455X: 192MB L2 >> working set).
// ---------------------------------------------------------------------------
__global__ void scatter_add(const float* __restrict__ xin,
                            const int* __restrict__ src,
                            const int* __restrict__ dst,
                            float* __restrict__ ssum, float* __restrict__ cnt,
                            int nE, int withCnt) {
  long i = (long)blockIdx.x * blockDim.x + threadIdx.x;
  long total = (long)nE * 16;
  long stride = (long)gridDim.x * blockDim.x;
  for (; i < total; i += stride) {
    int e = (int)(i >> 4);
    int q = (int)(i & 15);
    int s = src[e];
    int d = dst[e];
    const float4 v = *(const float4*)(xin + (long)s * D + q * 4);
    float* p = ssum + (long)d * D + q * 4;
    atomicAdd(p + 0, v.x);
    atomicAdd(p + 1, v.y);
    atomicAdd(p + 2, v.z);
    atomicAdd(p + 3, v.w);
    if (withCnt && q == 0) atomicAdd(cnt + d, 1.0f);
  }
}

// ---------------------------------------------------------------------------
// Fused mean + dual-GEMM + bias (+ReLU):
//   out[m0:m0+16, :] = (ssum/max(cnt,1)) @ Wl^T + xin @ Wr^T + b
// Block = 128 threads = 4 waves; wave w owns output columns [w*16, w*16+16).
// Uses native fp32 WMMA (V_WMMA_F32_16X16X4_F32) to keep reference precision.
// A-tile layout: lane l, vgpr j -> A[l%16][k0 + 2*(l/16) + j]
// B-tile layout: lane l, vgpr j -> B[k0 + 2*(l/16) + j][l%16]
// C/D layout:    lane l, vgpr i -> C[i + 8*(l/16)][l%16]
// ---------------------------------------------------------------------------
template <int RELU>
__global__ __launch_bounds__(128) void sage_gemm(
    const float* __restrict__ xin, const float* __restrict__ ssum,
    const float* __restrict__ cnt, const float* __restrict__ Wl,
    const float* __restrict__ Wr, const float* __restrict__ bias,
    float* __restrict__ out, int nN) {
  __shared__ float sMean[16 * D];
  __shared__ float sX[16 * D];
  const int m0 = blockIdx.x * 16;
  const int tid = threadIdx.x;

  // Stage mean and x tiles into LDS: 128 threads x 8 floats = 1024 per tile.
  {
    int idx = tid * 8;          // 8 consecutive floats, same row (64 % 8 == 0)
    int row = idx >> 6;         // tid / 8
    int col = idx & (D - 1);
    int gr = m0 + row;
    if (gr >= nN) gr = nN - 1;  // clamp (N is a multiple of 16 here anyway)
    float c = cnt[gr];
    float rinv = 1.0f / fmaxf(c, 1.0f);
    const float* ps = ssum + (long)gr * D + col;
    const float* px = xin + (long)gr * D + col;
#pragma unroll
    for (int t = 0; t < 8; ++t) {
      sMean[idx + t] = ps[t] * rinv;
      sX[idx + t] = px[t];
    }
  }
  __syncthreads();

  const int wave = tid >> 5;        // 0..3 -> output column tile
  const int lane = tid & 31;
  const int n0 = wave * 16;
  const int lr = lane & 15;
  const int kh = (lane >> 4) * 2;   // 0 or 2: K-half owned by this lane group

  const float* mrow = sMean + lr * D;               // A rows (mean)
  const float* xrow = sX + lr * D;                  // A rows (x)
  const float* wlrow = Wl + (n0 + lr) * D;          // B: Wl^T[k][n] = Wl[n][k]
  const float* wrrow = Wr + (n0 + lr) * D;          // B: Wr^T[k][n] = Wr[n][k]

  v8f acc = {};
#pragma unroll
  for (int k0 = 0; k0 < D; k0 += 4) {
    v2f a, b;
    // mean @ Wl^T
    a[0] = mrow[k0 + kh + 0];
    a[1] = mrow[k0 + kh + 1];
    b[0] = wlrow[k0 + kh + 0];
    b[1] = wlrow[k0 + kh + 1];
    acc = __builtin_amdgcn_wmma_f32_16x16x4_f32(false, a, false, b, (short)0,
                                                acc, false, false);
    // x @ Wr^T
    a[0] = xrow[k0 + kh + 0];
    a[1] = xrow[k0 + kh + 1];
    b[0] = wrrow[k0 + kh + 0];
    b[1] = wrrow[k0 + kh + 1];
    acc = __builtin_amdgcn_wmma_f32_16x16x4_f32(false, a, false, b, (short)0,
                                                acc, false, false);
  }

  const float bv = bias[n0 + lr];
  const int mbase = m0 + (lane >> 4) * 8;
#pragma unroll
  for (int i = 0; i < 8; ++i) {
    int r = mbase + i;
    if (r < nN) {
      float v = acc[i] + bv;
      if (RELU) v = fmaxf(v, 0.0f);
      out[(long)r * D + n0 + lr] = v;
    }
  }
}

// ---------------------------------------------------------------------------
// kernel_launch
// d_in: x[N*D] f32, edge_index[2*E] int, Wl1, Wr1, b1, Wl2, Wr2, b2 (f32)
// ws:   ssum[N*D] | cnt[N] | h[N*D]   (~25.8 MB)
// ---------------------------------------------------------------------------
extern "C" void kernel_launch(void* const* d_in, const int* in_sizes, int n_in,
                              void* d_out, int out_size, void* d_ws,
                              size_t ws_size, hipStream_t stream) {
  const float* x = (const float*)d_in[0];
  const int* edge = (const int*)d_in[1];
  const float* Wl1 = (const float*)d_in[2];
  const float* Wr1 = (const float*)d_in[3];
  const float* b1 = (const float*)d_in[4];
  const float* Wl2 = (const float*)d_in[5];
  const float* Wr2 = (const float*)d_in[6];
  const float* b2 = (const float*)d_in[7];

  const int N = in_sizes[0] / D;
  const int E = in_sizes[1] / 2;
  const int* src = edge;
  const int* dst = edge + E;

  float* ssum = (float*)d_ws;
  float* cnt = ssum + (long)N * D;
  float* h = cnt + N;
  float* out = (float*)d_out;

  const int gtiles = (N + 15) / 16;

  // Layer 1
  zero_f32<<<2048, 256, 0, stream>>>(ssum, (long)N * D);
  zero_f32<<<64, 256, 0, stream>>>(cnt, (long)N);
  scatter_add<<<4096, 256, 0, stream>>>(x, src, dst, ssum, cnt, E, 1);
  sage_gemm<1><<<gtiles, 128, 0, stream>>>(x, ssum, cnt, Wl1, Wr1, b1, h, N);

  // Layer 2 (degree counts identical -> reuse cnt)
  zero_f32<<<2048, 256, 0, stream>>>(ssum, (long)N * D);
  scatter_add<<<4096, 256, 0, stream>>>(h, src, dst, ssum, cnt, E, 0);
  sage_gemm<0><<<gtiles, 128, 0, stream>>>(h, ssum, cnt, Wl2, Wr2, b2, out, N);
}